// Group_44487271252314
// MI455X (gfx1250) — compile-verified
//
#include <hip/hip_runtime.h>
#include <hip/hip_bf16.h>

typedef __attribute__((ext_vector_type(2))) float v2f;
typedef __attribute__((ext_vector_type(8))) float v8f;

#define NPTS   8192
#define NGRP   512
#define KNN    32
#define NCH    6

// ---------------------------------------------------------------------------
// Kernel 1: farthest point sampling. One block per batch, 1024 threads,
// 8 points per thread held in registers. Sequential 512-step argmax loop.
// Writes centers to d_out tail region and (x,y,z,|c|^2) to workspace.
// ---------------------------------------------------------------------------
__global__ __launch_bounds__(1024) void fps_kernel(const float* __restrict__ pts,
                                                   float* __restrict__ centOut,
                                                   float4* __restrict__ wsC) {
    __shared__ float cent[3];
    __shared__ float wvals[32];
    __shared__ int   widx[32];
    __shared__ int   sFar;

    const int b    = blockIdx.x;
    const int t    = threadIdx.x;
    const int lane = t & 31;
    const int wid  = t >> 5;
    const long bOff = (long)b * NPTS;

    float x[8], y[8], z[8], dist[8];
#pragma unroll
    for (int j = 0; j < 8; ++j) {
        const int p = j * 1024 + t;
        const float* q = pts + (bOff + p) * NCH;
        x[j] = q[0]; y[j] = q[1]; z[j] = q[2];
        dist[j] = 1e10f;
    }

    int far = 0;
    for (int it = 0; it < NGRP; ++it) {
        if (t == 0) {
            const float* q = pts + (bOff + far) * NCH;  // L2-hot broadcast read
            cent[0] = q[0]; cent[1] = q[1]; cent[2] = q[2];
        }
        __syncthreads();
        const float cx = cent[0], cy = cent[1], cz = cent[2];
        if (t == 0) {
            const int o = b * NGRP + it;
            centOut[o * 3 + 0] = cx;
            centOut[o * 3 + 1] = cy;
            centOut[o * 3 + 2] = cz;
            wsC[o] = make_float4(cx, cy, cz, cx * cx + cy * cy + cz * cz);
        }
        float lv = -1.0f; int li = 0;
#pragma unroll
        for (int j = 0; j < 8; ++j) {
            const float dx = x[j] - cx, dy = y[j] - cy, dz = z[j] - cz;
            float d = dx * dx + dy * dy + dz * dz;
            d = fminf(dist[j], d);
            dist[j] = d;
            const int p = j * 1024 + t;
            if (d > lv || (d == lv && p < li)) { lv = d; li = p; }
        }
        // wave32 butterfly argmax (max value, lowest index on tie)
#pragma unroll
        for (int off = 16; off; off >>= 1) {
            const float ov = __shfl_xor(lv, off, 32);
            const int   oi = __shfl_xor(li, off, 32);
            if (ov > lv || (ov == lv && oi < li)) { lv = ov; li = oi; }
        }
        if (lane == 0) { wvals[wid] = lv; widx[wid] = li; }
        __syncthreads();
        if (wid == 0) {
            lv = wvals[lane]; li = widx[lane];
#pragma unroll
            for (int off = 16; off; off >>= 1) {
                const float ov = __shfl_xor(lv, off, 32);
                const int   oi = __shfl_xor(li, off, 32);
                if (ov > lv || (ov == lv && oi < li)) { lv = ov; li = oi; }
            }
            if (lane == 0) sFar = li;
        }
        __syncthreads();
        far = sFar;
    }
}

// ---------------------------------------------------------------------------
// Kernel 2: KNN + gather. One block per (batch, tile of 16 centers).
// 512 threads = 16 waves. Squared distances come straight out of
// V_WMMA_F32_16X16X4_F32: A = -2*center xyz (16x4, K padded), B = point xyz
// (4x16), C seeded with |c|^2 + |p|^2, so D is the 16x16 d2 tile.
// Points processed in 4 chunks of 2048 (128 KB LDS distance slab).
// Point loads are hoisted ahead of the WMMA batch (8 b96 loads in flight),
// and the next chunk is prefetched (global_prefetch_b8) before selection.
// Selection pulls each lane's 64-element strip into VGPRs once, then runs
// 32 masked min-extractions entirely in registers (no LDS re-reads).
// ---------------------------------------------------------------------------
__global__ __launch_bounds__(512) void knn_kernel(const float* __restrict__ pts,
                                                  const float4* __restrict__ wsC,
                                                  float* __restrict__ outNb) {
    extern __shared__ float smem[];
    float*  dch    = smem;                         // 16 * 2048 f32   (128 KB)
    float*  candd  = dch + 16 * 2048;              // 16 * 128  f32
    int*    candi  = (int*)(candd + 16 * 128);     // 16 * 128  i32
    float4* cts    = (float4*)(candi + 16 * 128);  // 16 float4 (16B aligned)
    int*    selidx = (int*)(cts + 16);             // 16 * 32   i32

    const int blk  = blockIdx.x;
    const int b    = blk >> 5;
    const int g0   = (blk & 31) << 4;
    const int t    = threadIdx.x;
    const int lane = t & 31;
    const int w    = t >> 5;
    const long bOff = (long)b * NPTS;

    if (t < 16) cts[t] = wsC[b * NGRP + g0 + t];
    __syncthreads();

    // A operand (constant across all tiles): -2 * center coords.
    // 16x4 f32 A layout: lanes 0-15 hold K0/K1 (x,y), lanes 16-31 hold K2/K3 (z,0).
    const float4 cmA = cts[lane & 15];
    v2f a;
    a.x = (lane < 16) ? (-2.0f * cmA.x) : (-2.0f * cmA.z);
    a.y = (lane < 16) ? (-2.0f * cmA.y) : 0.0f;

    // |c|^2 for the 8 C/D rows this lane-half owns (rows r + 8*(lane>=16)).
    float cn[8];
    {
        const int half = (lane >> 4) << 3;
#pragma unroll
        for (int r = 0; r < 8; ++r) cn[r] = cts[r + half].w;
    }

    for (int chunk = 0; chunk < 4; ++chunk) {
        const int base = chunk * 2048;

        // --- hoisted point loads: 8 global_load_b96 in flight before any use
        float px[8], py[8], pz[8];
#pragma unroll
        for (int j = 0; j < 8; ++j) {
            const int n = base + (w + j * 16) * 16 + (lane & 15);
            const float* q = pts + (bOff + n) * NCH;
            px[j] = q[0]; py[j] = q[1]; pz[j] = q[2];
        }
        // --- prefetch next chunk's point rows into WGP$/L2
        if (chunk < 3) {
#pragma unroll
            for (int j = 0; j < 8; ++j) {
                const int n = base + 2048 + (w + j * 16) * 16 + (lane & 15);
                __builtin_prefetch(pts + (bOff + n) * NCH, 0, 1);
            }
        }

        // --- distance tiles via WMMA: wave w covers point-tiles w, w+16, ...
#pragma unroll
        for (int j = 0; j < 8; ++j) {
            const int tile = w + j * 16;
            const float pn2 = px[j] * px[j] + py[j] * py[j] + pz[j] * pz[j];
            // 4x16 f32 B layout: lanes 0-15 rows K0/K1 (x,y), 16-31 rows K2/K3 (z,0)
            v2f bb;
            bb.x = (lane < 16) ? px[j] : pz[j];
            bb.y = (lane < 16) ? py[j] : 0.0f;
            v8f c0;
#pragma unroll
            for (int r = 0; r < 8; ++r) c0[r] = cn[r] + pn2;
            // D = (-2c).p + (|c|^2 + |p|^2) = squared distance tile
            v8f d = __builtin_amdgcn_wmma_f32_16x16x4_f32(
                false, a, false, bb, (short)0, c0, false, false);
            const int half = (lane >> 4) << 3;
            const int col  = tile * 16 + (lane & 15);
#pragma unroll
            for (int r = 0; r < 8; ++r) dch[(r + half) * 2048 + col] = d[r];
        }
        __syncthreads();

        // --- pull this lane's 64-element strip of center-row w into VGPRs
        float rv[64];
#pragma unroll
        for (int j = 0; j < 64; ++j) rv[j] = dch[w * 2048 + lane + (j << 5)];

        // --- 32 min-extractions, entirely register-resident (taken-bitmask)
        unsigned long long mask = 0ull;
        for (int it = 0; it < KNN; ++it) {
            float lv = 3.4e38f; int li = 0x7fffffff;
#pragma unroll
            for (int j = 0; j < 64; ++j) {
                const float v = (mask & (1ull << j)) ? 3.4e38f : rv[j];
                const int gi = base + lane + (j << 5);
                if (v < lv || (v == lv && gi < li)) { lv = v; li = gi; }
            }
#pragma unroll
            for (int off = 16; off; off >>= 1) {
                const float ov = __shfl_xor(lv, off, 32);
                const int   oi = __shfl_xor(li, off, 32);
                if (ov < lv || (ov == lv && oi < li)) { lv = ov; li = oi; }
            }
            const int col = li - base;
            if (lane == (col & 31)) mask |= (1ull << (col >> 5));
            if (lane == 0) {
                candd[w * 128 + chunk * KNN + it] = lv;
                candi[w * 128 + chunk * KNN + it] = li;
            }
        }
        __syncthreads();  // before the distance slab is overwritten
    }

    // --- merge 4x32 per-chunk candidates into the final ascending 32
    {
        unsigned int mask = 0u;
        for (int it = 0; it < KNN; ++it) {
            float lv = 3.4e38f; int li = 0x7fffffff; int lp = 0;
#pragma unroll
            for (int j = 0; j < 4; ++j) {
                if (mask & (1u << j)) continue;
                const int pos = lane + (j << 5);
                const float v = candd[w * 128 + pos];
                const int gi = candi[w * 128 + pos];
                if (v < lv || (v == lv && gi < li)) { lv = v; li = gi; lp = pos; }
            }
#pragma unroll
            for (int off = 16; off; off >>= 1) {
                const float ov = __shfl_xor(lv, off, 32);
                const int   oi = __shfl_xor(li, off, 32);
                const int   op = __shfl_xor(lp, off, 32);
                if (ov < lv || (ov == lv && oi < li)) { lv = ov; li = oi; lp = op; }
            }
            if (lane == (lp & 31)) mask |= (1u << (lp >> 5));
            if (lane == 0) selidx[w * KNN + it] = li;
        }
    }
    __syncthreads();

    // --- gather: wave w handles center w, lane handles neighbor #lane
    {
        const int gidx = selidx[w * KNN + lane];
        const float4 cm = cts[w];
        const float* q = pts + (bOff + gidx) * NCH;
        float* o = outNb + (((long)(b * NGRP + g0 + w) * KNN + lane) * NCH);
        o[0] = q[0] - cm.x;
        o[1] = q[1] - cm.y;
        o[2] = q[2] - cm.z;
        o[3] = q[3];
        o[4] = q[4];
        o[5] = q[5];
    }
}

extern "C" void kernel_launch(void* const* d_in, const int* in_sizes, int n_in,
                              void* d_out, int out_size, void* d_ws, size_t ws_size,
                              hipStream_t stream) {
    const float* pts = (const float*)d_in[0];
    float* out = (float*)d_out;
    float* centOut = out + (size_t)8 * NGRP * KNN * NCH;  // center region tail
    float4* wsC = (float4*)d_ws;                           // 8*512 float4 = 64 KB

    fps_kernel<<<8, 1024, 0, stream>>>(pts, centOut, wsC);

    const size_t smemBytes =
        (size_t)(16 * 2048) * sizeof(float) +   // distance slab
        (size_t)(16 * 128) * sizeof(float) +    // candidate values
        (size_t)(16 * 128) * sizeof(int) +      // candidate indices
        (size_t)16 * sizeof(float4) +           // center tile
        (size_t)(16 * KNN) * sizeof(int);       // final indices
    knn_kernel<<<8 * 32, 512, smemBytes, stream>>>(pts, wsC, out);
}